// LightGCN_3015067042276
// MI455X (gfx1250) — compile-verified
//
#include <hip/hip_runtime.h>
#include <hip/hip_bf16.h>

typedef __attribute__((ext_vector_type(2))) float v2f;
typedef __attribute__((ext_vector_type(8))) float v8f;

#define D 64

// ---------------------------------------------------------------------------
// Degree / normalization
// ---------------------------------------------------------------------------
__global__ void deg_init_kernel(float* deg, int n) {
    int i = blockIdx.x * blockDim.x + threadIdx.x;
    if (i < n) deg[i] = 1.0f;  // self-loop contributes 1 to every node's degree
}

__global__ void deg_count_kernel(const long long* __restrict__ col, float* deg, int e) {
    int i = blockIdx.x * blockDim.x + threadIdx.x;
    if (i < e) atomicAdd(&deg[(int)col[i]], 1.0f);
}

__global__ void dinv_kernel(float* deg, int n) {
    int i = blockIdx.x * blockDim.x + threadIdx.x;
    if (i < n) deg[i] = rsqrtf(deg[i]);  // deg >= 1 always (self-loop)
}

// ---------------------------------------------------------------------------
// y = x @ W  via V_WMMA_F32_16X16X4_F32.
// One wave -> 16 rows x 64 cols (4 accumulator tiles), K-loop of 16 steps.
// x is split across two pointers (user_emb | item_emb) for layer 1.
// ---------------------------------------------------------------------------
__global__ void __launch_bounds__(256)
gemm_xw_kernel(const float* __restrict__ xa, const float* __restrict__ xb,
               int splitRow, const float* __restrict__ W,
               float* __restrict__ y, int nrows) {
    __shared__ float Ws[D * D];  // 16 KB
    // Cooperative load of W into LDS (float4 = 4 iterations per thread @ 256 thr)
    for (int i = threadIdx.x; i < (D * D) / 4; i += blockDim.x)
        ((float4*)Ws)[i] = ((const float4*)W)[i];
    __syncthreads();

    const int wave = threadIdx.x >> 5;
    const int lane = threadIdx.x & 31;
    const int rowTile = (blockIdx.x * (blockDim.x >> 5) + wave) * 16;
    if (rowTile >= nrows) return;  // wave-uniform: EXEC stays all-ones for WMMA

    const int m  = lane & 15;   // row within tile (A) / col within tile (B,C,D)
    const int hi = lane >> 4;   // K-pair selector for A/B fragments

    const int r = rowTile + m;
    const float* __restrict__ xr =
        (r < splitRow) ? (xa + (size_t)r * D) : (xb + (size_t)(r - splitRow) * D);

    v8f acc0 = {}, acc1 = {}, acc2 = {}, acc3 = {};

    #pragma unroll
    for (int k0 = 0; k0 < D; k0 += 4) {
        // A fragment (16x4 f32): lanes 0-15 -> K = k0,k0+1 ; lanes 16-31 -> K = k0+2,k0+3
        v2f a;
        a.x = xr[k0 + 2 * hi + 0];
        a.y = xr[k0 + 2 * hi + 1];
        // B fragments (4x16 f32) for the 4 column tiles, from LDS
        #pragma unroll
        for (int ct = 0; ct < 4; ++ct) {
            const int n = ct * 16 + m;
            v2f b;
            b.x = Ws[(k0 + 2 * hi + 0) * D + n];
            b.y = Ws[(k0 + 2 * hi + 1) * D + n];
            v8f c = (ct == 0) ? acc0 : (ct == 1) ? acc1 : (ct == 2) ? acc2 : acc3;
            c = __builtin_amdgcn_wmma_f32_16x16x4_f32(
                    /*neg_a=*/false, a, /*neg_b=*/false, b,
                    /*c_mod=*/(short)0, c, /*reuse_a=*/false, /*reuse_b=*/false);
            if (ct == 0) acc0 = c; else if (ct == 1) acc1 = c;
            else if (ct == 2) acc2 = c; else acc3 = c;
        }
    }

    // C/D layout: VGPR r -> row (r + 8*hi), lane&15 -> col
    #pragma unroll
    for (int rr = 0; rr < 8; ++rr) {
        const size_t row = (size_t)(rowTile + rr + 8 * hi);
        y[row * D +  0 + m] = acc0[rr];
        y[row * D + 16 + m] = acc1[rr];
        y[row * D + 32 + m] = acc2[rr];
        y[row * D + 48 + m] = acc3[rr];
    }
}

// ---------------------------------------------------------------------------
// Self-loop + bias init: out[n,:] = y[n,:] * dinv[n]^2 + b   (fully writes out)
// One thread per float2 (n*32 threads).
// ---------------------------------------------------------------------------
__global__ void agg_init_kernel(const float* __restrict__ y,
                                const float* __restrict__ dinv,
                                const float* __restrict__ b,
                                float* __restrict__ out, int n) {
    int tid = blockIdx.x * blockDim.x + threadIdx.x;
    if (tid >= n * 32) return;
    const int node = tid >> 5;
    const int p    = tid & 31;
    const float di = dinv[node];
    const float s  = di * di;
    const float2 v = ((const float2*)(y + (size_t)node * D))[p];
    float2 o;
    o.x = v.x * s + b[2 * p + 0];
    o.y = v.y * s + b[2 * p + 1];
    ((float2*)(out + (size_t)node * D))[p] = o;
}

// ---------------------------------------------------------------------------
// Edge scatter: out[col,:] += y[row,:] * dinv[row]*dinv[col]
// One wave per edge; lane p handles dims {2p, 2p+1}. L2 f32 atomics.
// ---------------------------------------------------------------------------
__global__ void agg_edges_kernel(const long long* __restrict__ ei,
                                 const float* __restrict__ y,
                                 const float* __restrict__ dinv,
                                 float* __restrict__ out, int E) {
    int tid = blockIdx.x * blockDim.x + threadIdx.x;
    if (tid >= E * 32) return;
    const int e = tid >> 5;
    const int p = tid & 31;
    const int r = (int)ei[e];
    const int c = (int)ei[(size_t)E + e];
    const float norm = dinv[r] * dinv[c];
    const float2 v = ((const float2*)(y + (size_t)r * D))[p];
    atomicAdd(&out[(size_t)c * D + 2 * p + 0], v.x * norm);
    atomicAdd(&out[(size_t)c * D + 2 * p + 1], v.y * norm);
}

// ---------------------------------------------------------------------------
extern "C" void kernel_launch(void* const* d_in, const int* in_sizes, int n_in,
                              void* d_out, int out_size, void* d_ws, size_t ws_size,
                              hipStream_t stream) {
    const long long* edge = (const long long*)d_in[0];  // [2, E] int64
    const float* user = (const float*)d_in[1];          // [NU, 64]
    const float* item = (const float*)d_in[2];          // [NI, 64]
    const float* W1   = (const float*)d_in[3];
    const float* b1   = (const float*)d_in[4];
    const float* W2   = (const float*)d_in[5];
    const float* b2   = (const float*)d_in[6];
    float* out = (float*)d_out;

    const int E  = in_sizes[0] / 2;
    const int NU = in_sizes[1] / D;
    const int NI = in_sizes[2] / D;
    const int N  = NU + NI;

    // Workspace layout: dinv [N] | y [N*64] | x1 [N*64]   (~103 MB)
    float* dinv = (float*)d_ws;
    float* y    = dinv + N;
    float* x1   = y + (size_t)N * D;

    const int T = 256;
    const int gN    = (N + T - 1) / T;
    const int gE    = (E + T - 1) / T;
    const int gElem = (N * 32 + T - 1) / T;       // N*64 floats as float2
    const int gEdge = ((size_t)E * 32 + T - 1) / T;
    const int wavesPerBlock = T / 32;
    const int gGemm = ((N + 15) / 16 + wavesPerBlock - 1) / wavesPerBlock;

    // 1) Normalization: deg -> dinv
    deg_init_kernel<<<gN, T, 0, stream>>>(dinv, N);
    deg_count_kernel<<<gE, T, 0, stream>>>(edge + (size_t)E, dinv, E);
    dinv_kernel<<<gN, T, 0, stream>>>(dinv, N);

    // 2) Layer 1: y = [user;item] @ W1 ; x1 = scatter(y) + b1
    gemm_xw_kernel<<<gGemm, T, 0, stream>>>(user, item, NU, W1, y, N);
    agg_init_kernel<<<gElem, T, 0, stream>>>(y, dinv, b1, x1, N);
    agg_edges_kernel<<<gEdge, T, 0, stream>>>(edge, y, dinv, x1, E);

    // 3) Layer 2: y = x1 @ W2 ; out = scatter(y) + b2   (reuse y buffer)
    gemm_xw_kernel<<<gGemm, T, 0, stream>>>(x1, x1, N, W2, y, N);
    agg_init_kernel<<<gElem, T, 0, stream>>>(y, dinv, b2, out, N);
    agg_edges_kernel<<<gEdge, T, 0, stream>>>(edge, y, dinv, out, E);
}